// GeneSetAttentionAggregator_72782515798446
// MI455X (gfx1250) — compile-verified
//
#include <hip/hip_runtime.h>
#include <cstdint>
#include <cstddef>

// Problem constants (from the reference).
namespace {
constexpr int kB = 64;      // batch
constexpr int kG = 20000;   // genes
constexpr int kD = 64;      // feature dim
constexpr int kS = 512;     // sets
constexpr int kL = 128;     // max set length
constexpr int kWavesB = 4;  // waves per aggregate block (one b per wave)
constexpr int kStride = 80; // LDS tile row stride in floats (64 + 16 TDM pad)
constexpr int kTileBytes = 16 * kStride * 4;  // 5120B per 16-row tile
}

typedef __attribute__((ext_vector_type(4))) unsigned int u32x4;
typedef __attribute__((ext_vector_type(8))) int          i32x8;
typedef __attribute__((ext_vector_type(4))) int          i32x4;
typedef __attribute__((ext_vector_type(2))) float        v2f;
typedef __attribute__((ext_vector_type(8))) float        v8f;

// ---------------------------------------------------------------------------
// Kernel 1: masked softmax over each attn_logits row, plus mask popcount.
// ---------------------------------------------------------------------------
__global__ __launch_bounds__(kL) void masked_softmax_kernel(
    const float* __restrict__ logits,
    const unsigned char* __restrict__ mask,   // jnp.bool_ -> 1 byte
    float* __restrict__ w_out,
    int* __restrict__ len_out) {
  __shared__ float red[kL];
  const int s = blockIdx.x;
  const int t = threadIdx.x;
  const int base = s * kL + t;

  const bool m = mask[base] != 0;
  const float x = m ? logits[base] : -3.402823466e38f;

  red[t] = x;
  __syncthreads();
#pragma unroll
  for (int off = kL / 2; off > 0; off >>= 1) {
    if (t < off) red[t] = fmaxf(red[t], red[t + off]);
    __syncthreads();
  }
  const float mx = red[0];
  __syncthreads();

  const float e = m ? __expf(x - mx) : 0.0f;
  red[t] = e;
  __syncthreads();
#pragma unroll
  for (int off = kL / 2; off > 0; off >>= 1) {
    if (t < off) red[t] += red[t + off];
    __syncthreads();
  }
  const float denom = red[0];
  __syncthreads();

  w_out[base] = e / denom;   // exact 0 on masked lanes (incl. all l >= L)

  red[t] = m ? 1.0f : 0.0f;
  __syncthreads();
#pragma unroll
  for (int off = kL / 2; off > 0; off >>= 1) {
    if (t < off) red[t] += red[t + off];
    __syncthreads();
  }
  if (t == 0) len_out[s] = (int)red[0];
}

// ---------------------------------------------------------------------------
// TDM gather-mode descriptor: fetch `nrows` indexed rows (16-bit indices from
// LDS s_idx[c*16..]) of the 20000x64 f32 tensor at `gaddr` into the LDS tile
// at `lds_tile`, padding each 64-DWORD row by 16 DWORDs (stride 80) so the
// WMMA B-operand reads are LDS-bank-conflict free.
// ---------------------------------------------------------------------------
__device__ __forceinline__ void tdm_issue_gather(
    const int* __restrict__ sidx,   // LDS idx row (128 entries)
    int c, int nrows, unsigned lds_tile, uint64_t gaddr) {
  const int c16 = c << 4;
  unsigned p[8];
#pragma unroll
  for (int j = 0; j < 8; ++j) {
    const unsigned lo = (unsigned)sidx[c16 + 2 * j] & 0xFFFFu;
    const unsigned hi = (unsigned)sidx[c16 + 2 * j + 1] & 0xFFFFu;
    p[j] = (unsigned)__builtin_amdgcn_readfirstlane((int)(lo | (hi << 16)));
  }

  // D# group 0 (§8.3): count=1, gather_mode=1, 16-bit indices, lds_addr,
  // 57-bit global_addr, type=2.
  u32x4 g0;
  g0[0] = 0x80000001u;
  g0[1] = (unsigned)__builtin_amdgcn_readfirstlane((int)lds_tile);
  g0[2] = (unsigned)__builtin_amdgcn_readfirstlane((int)(unsigned)gaddr);
  g0[3] = (unsigned)__builtin_amdgcn_readfirstlane(
              (int)((((unsigned)(gaddr >> 32)) & 0x01FFFFFFu) | (2u << 30)));

  // D# group 1 (§8.4): data_size=2 (4B) | pad_enable | pad_interval=5
  // (every 64 DWORDs) | pad_amount=15 (+16 DWORDs); tensor 20000x64,
  // tile nrows x 64, dim0 stride 64.
  i32x8 g1;
  g1[0] = (int)0x1F520000;            // pads + 4B elements
  g1[1] = (int)(64u << 16);           // tensor_dim0 at bits[79:48]
  g1[2] = (int)(20000u << 16);        // tensor_dim1 at bits[111:80]
  g1[3] = (int)(64u << 16);           // tile_dim0 at bits[127:112]
  g1[4] = __builtin_amdgcn_readfirstlane(nrows & 0xFFFF);  // tile_dim1
  g1[5] = 64;                         // tensor_dim0_stride (48b, low word)
  g1[6] = 0;
  g1[7] = 0;

  // Groups 2/3 (§8.7): 16 packed 16-bit row indices.
  i32x4 g2 = { (int)p[0], (int)p[1], (int)p[2], (int)p[3] };
  i32x4 g3 = { (int)p[4], (int)p[5], (int)p[6], (int)p[7] };

  i32x8 gx = { 0, 0, 0, 0, 0, 0, 0, 0 };   // unused extra group
  __builtin_amdgcn_tensor_load_to_lds(g0, g1, g2, g3, gx, 0);
}

// ---------------------------------------------------------------------------
// WMMA consume of one 16x64 LDS tile:
//   acc[nt](16x16) += A(16x4, weight rows replicated) x B(4x16 tile slab)
// A layout (f32 16x4, §7.12.2): lanes 0-15 hold K=0(V0),1(V1); lanes 16-31
// hold K=2,3. B mirrored: lanes 0-15 = N, rows k..k+1; lanes 16-31 rows
// k+2..k+3. Weights are 0 for l >= L so ragged chunks need no special case.
// ---------------------------------------------------------------------------
__device__ __forceinline__ void wmma_consume_chunk(
    const float* tile, const float* sw, int c, int lane, v8f acc[4]) {
  const int  c16  = c << 4;
  const int  hsel = (lane < 16) ? 0 : 2;
  v2f a[4];
#pragma unroll
  for (int k = 0; k < 4; ++k) {
    const int k0 = c16 + (k << 2);
    a[k].x = sw[k0 + hsel];
    a[k].y = sw[k0 + hsel + 1];
  }
  const int colb = lane & 15;
#pragma unroll
  for (int nt = 0; nt < 4; ++nt) {
#pragma unroll
    for (int k = 0; k < 4; ++k) {
      const int r0 = (k << 2) + hsel;
      v2f bm;
      bm.x = tile[(r0 + 0) * kStride + nt * 16 + colb];
      bm.y = tile[(r0 + 1) * kStride + nt * 16 + colb];
      acc[nt] = __builtin_amdgcn_wmma_f32_16x16x4_f32(
          false, a[k], false, bm, (short)0, acc[nt], false, false);
    }
  }
}

// ---------------------------------------------------------------------------
// Kernel 2: out[b,s,:] = sum_l w[s,l] * gene[b, idx[s,l], :]
// Block = 4 wave32; wave w owns b = bg*4 + w for set s. TDM gathers 16
// indexed rows per descriptor into a double-buffered padded LDS tile per
// wave (TENSORcnt pipelining); the weighted reduction of each tile runs on
// the XDL via replicated-A V_WMMA_F32_16X16X4_F32. Grid is b-slab-major so
// each slab's gathered working set (~34MB) stays resident in the 192MB L2.
// ---------------------------------------------------------------------------
__global__ __launch_bounds__(32 * kWavesB) void aggregate_kernel(
    const float* __restrict__ gene,
    const int* __restrict__ idx,
    const float* __restrict__ w,
    const int* __restrict__ len,
    float* __restrict__ out) {
  __shared__ __align__(16) float tiles[kWavesB][2][16 * kStride];  // 40KB
  __shared__ __align__(16) int   s_idx[kL];
  __shared__ __align__(16) float s_w[kL];

  const int s    = blockIdx.x % kS;        // slab-major ordering
  const int bg   = blockIdx.x / kS;
  const int wave = threadIdx.x >> 5;
  const int lane = threadIdx.x & 31;
  const int b    = bg * kWavesB + wave;

  // --- async staging of idx + weight rows into LDS (wave 0 only) ---
  if (threadIdx.x < 32) {
    unsigned lds_i = (unsigned)(uintptr_t)(&s_idx[0]) + (unsigned)lane * 16u;
    unsigned lds_w = (unsigned)(uintptr_t)(&s_w[0])   + (unsigned)lane * 16u;
    const char* gi = (const char*)(idx + (size_t)s * kL) + lane * 16;
    const char* gw = (const char*)(w   + (size_t)s * kL) + lane * 16;
    asm volatile("global_load_async_to_lds_b128 %0, %1, off"
                 :: "v"(lds_i), "v"(gi) : "memory");
    asm volatile("global_load_async_to_lds_b128 %0, %1, off"
                 :: "v"(lds_w), "v"(gw) : "memory");
  }
  const int L = len[s];                      // uniform -> scalar load
  asm volatile("s_wait_asynccnt 0x0" ::: "memory");
  __syncthreads();

  const uint64_t gaddr =
      (uint64_t)(uintptr_t)gene + (uint64_t)b * ((uint64_t)kG * kD * 4u);
  const unsigned lds_base = (unsigned)(uintptr_t)(&tiles[wave][0][0]);

  const int nfull  = L >> 4;
  const int rem    = L & 15;
  const int nchunk = nfull + (rem ? 1 : 0);

  v8f acc[4];
#pragma unroll
  for (int nt = 0; nt < 4; ++nt)
    acc[nt] = (v8f){0.f, 0.f, 0.f, 0.f, 0.f, 0.f, 0.f, 0.f};

  // Prime the pipeline with chunk 0.
  tdm_issue_gather(s_idx, 0, (0 < nfull) ? 16 : rem, lds_base, gaddr);

  for (int c = 0; c < nchunk - 1; ++c) {
    // Prefetch next chunk while current is in flight.
    tdm_issue_gather(s_idx, c + 1, (c + 1 < nfull) ? 16 : rem,
                     lds_base + (unsigned)(((c + 1) & 1) * kTileBytes), gaddr);
    __builtin_amdgcn_s_wait_tensorcnt(1);    // chunk c complete
    wmma_consume_chunk(&tiles[wave][c & 1][0], s_w, c, lane, acc);
  }

  // Drain the last chunk (zero weights neutralize rows >= rem).
  __builtin_amdgcn_s_wait_tensorcnt(0);
  wmma_consume_chunk(&tiles[wave][(nchunk - 1) & 1][0], s_w, nchunk - 1, lane,
                     acc);

  // D-matrix row M=0: VGPR0, lanes 0-15 hold N=0..15 of each 16-d tile.
  if (lane < 16) {
    float* ob = out + ((size_t)b * kS + s) * kD;
#pragma unroll
    for (int nt = 0; nt < 4; ++nt) ob[nt * 16 + lane] = acc[nt][0];
  }
}

// ---------------------------------------------------------------------------
extern "C" void kernel_launch(void* const* d_in, const int* in_sizes, int n_in,
                              void* d_out, int out_size, void* d_ws, size_t ws_size,
                              hipStream_t stream) {
  const float*         gene   = (const float*)d_in[0];         // (B,G,D) f32
  const int*           idx    = (const int*)d_in[1];           // (S,LMAX) i32
  const unsigned char* mask   = (const unsigned char*)d_in[2]; // (S,LMAX) bool
  const float*         logits = (const float*)d_in[3];         // (S,LMAX) f32

  float* w_tab  = (float*)d_ws;                                  // S*LMAX f32
  int*   len_ws = (int*)((char*)d_ws + (size_t)kS * kL * sizeof(float)); // S i32

  masked_softmax_kernel<<<kS, kL, 0, stream>>>(logits, mask, w_tab, len_ws);

  aggregate_kernel<<<kS * (kB / kWavesB), 32 * kWavesB, 0, stream>>>(
      gene, idx, w_tab, len_ws, (float*)d_out);
}